// GIN_39788577030305
// MI455X (gfx1250) — compile-verified
//
#include <hip/hip_runtime.h>

typedef float v2f __attribute__((ext_vector_type(2)));
typedef float v8f __attribute__((ext_vector_type(8)));

#define DF 96
#define NT 6            // 96 / 16 column tiles
#define KP 48           // 96 / 2 k-pairs
#define LSTR 112        // padded LDS row stride in float2 (224 dwords == 32 mod 64)
#define BN_EPS 1e-5f

// ---------------------------------------------------------------- utilities
__global__ void copy_f4(float4* __restrict__ dst, const float4* __restrict__ src, int n4) {
    int i = blockIdx.x * blockDim.x + threadIdx.x;
    if (i < n4) dst[i] = src[i];
}

__global__ void zero_f32(float* __restrict__ p, int n) {
    int i = blockIdx.x * blockDim.x + threadIdx.x;
    if (i < n) p[i] = 0.f;
}

// ------------------------------------------------- edge scatter: agg += h[src]
// one thread = (edge, 4-feature chunk); E*24 threads total
__global__ void scatter_add(float* __restrict__ agg, const float* __restrict__ h,
                            const int* __restrict__ src, const int* __restrict__ dst, int E) {
    int idx = blockIdx.x * blockDim.x + threadIdx.x;
    int e = idx / 24, c = idx % 24;
    if (e >= E) return;
    const float4 v = ((const float4*)(h + (long)src[e] * DF))[c];
    float* b = agg + (long)dst[e] * DF + c * 4;
    atomicAdd(b + 0, v.x);
    atomicAdd(b + 1, v.y);
    atomicAdd(b + 2, v.z);
    atomicAdd(b + 3, v.w);
}

// stage W[96,96] into LDS as [kpair][col] float2 pairs (padded rows)
__device__ __forceinline__ void stage_w(float2* lw2, const float* __restrict__ W,
                                        int tid, int nthreads) {
    for (int i = tid; i < KP * DF; i += nthreads) {
        int kp = i / DF, col = i - kp * DF;           // coalesced over col
        float2 p;
        p.x = W[(2 * kp)     * DF + col];
        p.y = W[(2 * kp + 1) * DF + col];
        lw2[kp * LSTR + col] = p;
    }
}

// ---------------------------------------------- Y[M,96] = X[M,96]*W[96,96]+b
// block = 256 threads = 8 waves; wave owns one 16-row tile, all 6 col tiles.
__global__ void __launch_bounds__(256)
gemm_wmma_f32(float* __restrict__ Y, const float* __restrict__ X,
              const float* __restrict__ W, const float* __restrict__ bias, int M) {
    __shared__ __align__(16) float2 lw2[KP * LSTR];
    stage_w(lw2, W, threadIdx.x, 256);
    __syncthreads();

    const int wave = threadIdx.x >> 5;
    const int lane = threadIdx.x & 31;
    const int l15  = lane & 15;          // A: row-in-tile, B/D: col-in-tile
    const int kh   = lane >> 4;          // selects K pair {0,1} vs {2,3}
    const int nMT  = M / 16;             // M is a multiple of 16 (50000, 64)
    int t = blockIdx.x * 8 + wave;
    const bool active = t < nMT;
    if (!active) t = nMT - 1;            // clamp: redundant work, uniform barriers

    const float* xr = X + (long)(t * 16 + l15) * DF;

    v8f acc[NT] = {};
    for (int k = 0; k < DF; k += 4) {
        const float2 av = *(const float2*)(xr + k + 2 * kh);
        v2f a; a.x = av.x; a.y = av.y;
        const float2* brow = lw2 + (k / 2 + kh) * LSTR + l15;
#pragma unroll
        for (int n = 0; n < NT; ++n) {
            v2f b = *(const v2f*)(brow + n * 16);     // single ds_load_b64
            acc[n] = __builtin_amdgcn_wmma_f32_16x16x4_f32(
                false, a, false, b, (short)0, acc[n], false, false);
        }
    }
    if (!active) return;
    const int mBase = t * 16;
#pragma unroll
    for (int n = 0; n < NT; ++n) {
        const float bv = bias[n * 16 + l15];
#pragma unroll
        for (int r = 0; r < 8; ++r) {
            int m = mBase + r + 8 * kh;  // C layout: VGPR r -> M=r (+8 for hi half)
            Y[(long)m * DF + n * 16 + l15] = acc[n][r] + bv;
        }
    }
}

// ------------------------------------- per-column sum / sumsq (96 threads/blk)
__global__ void col_stats(const float* __restrict__ X, float* __restrict__ stats, int M) {
    int c  = threadIdx.x;
    int r0 = blockIdx.x * 256;
    int r1 = r0 + 256; if (r1 > M) r1 = M;
    float s = 0.f, q = 0.f;
    for (int r = r0; r < r1; ++r) {
        float x = X[(long)r * DF + c];
        s += x; q += x * x;
    }
    atomicAdd(&stats[c], s);
    atomicAdd(&stats[DF + c], q);
}

__global__ void finalize_stats(float* __restrict__ stats, const float* __restrict__ g,
                               const float* __restrict__ b, int M) {
    int c = threadIdx.x;
    float mean = stats[c] / (float)M;
    float var  = stats[DF + c] / (float)M - mean * mean;
    float sc   = g[c] * rsqrtf(var + BN_EPS);
    stats[c]      = sc;                  // scale
    stats[DF + c] = b[c] - mean * sc;    // shift
}

__global__ void bn_relu(float* __restrict__ X, const float* __restrict__ stats, int n) {
    int i = blockIdx.x * blockDim.x + threadIdx.x;
    if (i >= n) return;
    int c = i % DF;
    float z = stats[c] * X[i] + stats[DF + c];
    X[i] = z > 0.f ? z : 0.f;
}

__global__ void bn_prelu(float* __restrict__ O, const float* __restrict__ X,
                         const float* __restrict__ stats, const float* __restrict__ pa, int n) {
    int i = blockIdx.x * blockDim.x + threadIdx.x;
    if (i >= n) return;
    int c = i % DF;
    float z = stats[c] * X[i] + stats[DF + c];
    O[i] = z >= 0.f ? z : pa[0] * z;
}

// --------------------- graph max-pool; deterministic contiguous segment bounds
__global__ void pool_max(float* __restrict__ pooled, const float* __restrict__ H,
                         int M, int G) {
    int g = blockIdx.x, d = threadIdx.x;
    long lo = ((long)g * M + G - 1) / G;         // first i with (i*G)/M == g
    long hi = ((long)(g + 1) * M + G - 1) / G;
    float m = -3.402823466e38f;
    for (long i = lo; i < hi; ++i) m = fmaxf(m, H[i * DF + d]);
    pooled[g * DF + d] = m;
}

// --------------- heads: out[b, d*5 + l] = (pooled_l @ linW_l + linb_l)[b, d]
__global__ void __launch_bounds__(128)
head_gemm(float* __restrict__ out, const float* __restrict__ pooled,
          const float* __restrict__ linW, const float* __restrict__ linb,
          int G, int nHeads) {
    __shared__ __align__(16) float2 lw2[KP * LSTR];
    const int head = blockIdx.x;
    stage_w(lw2, linW + (long)head * DF * DF, threadIdx.x, 128);
    __syncthreads();

    const int wave = threadIdx.x >> 5;   // G=64 -> 4 row tiles = 4 waves
    const int lane = threadIdx.x & 31;
    const int l15  = lane & 15;
    const int kh   = lane >> 4;
    const float* xr = pooled + (long)head * G * DF + (long)(wave * 16 + l15) * DF;

    v8f acc[NT] = {};
    for (int k = 0; k < DF; k += 4) {
        const float2 av = *(const float2*)(xr + k + 2 * kh);
        v2f a; a.x = av.x; a.y = av.y;
        const float2* brow = lw2 + (k / 2 + kh) * LSTR + l15;
#pragma unroll
        for (int n = 0; n < NT; ++n) {
            v2f b = *(const v2f*)(brow + n * 16);
            acc[n] = __builtin_amdgcn_wmma_f32_16x16x4_f32(
                false, a, false, b, (short)0, acc[n], false, false);
        }
    }
    const int stride = DF * nHeads;      // 480
#pragma unroll
    for (int n = 0; n < NT; ++n) {
        int col = n * 16 + l15;
        float bv = linb[head * DF + col];
#pragma unroll
        for (int r = 0; r < 8; ++r) {
            int m = wave * 16 + r + 8 * kh;
            out[(long)m * stride + col * nHeads + head] = acc[n][r] + bv;
        }
    }
}

// --------------------------------------------------------------------- driver
extern "C" void kernel_launch(void* const* d_in, const int* in_sizes, int n_in,
                              void* d_out, int out_size, void* d_ws, size_t ws_size,
                              hipStream_t stream) {
    const float* h_in  = (const float*)d_in[0];
    const float* gW1   = (const float*)d_in[1];
    const float* gb1   = (const float*)d_in[2];
    const float* gbn1g = (const float*)d_in[3];
    const float* gbn1b = (const float*)d_in[4];
    const float* gW2   = (const float*)d_in[5];
    const float* gb2   = (const float*)d_in[6];
    const float* bng   = (const float*)d_in[7];
    const float* bnb   = (const float*)d_in[8];
    const float* pa    = (const float*)d_in[9];
    const float* linW  = (const float*)d_in[10];
    const float* linb  = (const float*)d_in[11];
    const int*   srcI  = (const int*)d_in[12];
    const int*   dstI  = (const int*)d_in[13];

    const int M = in_sizes[0] / DF;      // 50000
    const int E = in_sizes[12];          // 800000
    const int G = 64, NL = 4, NH = 5;
    const long nd = (long)M * DF;

    float* A      = (float*)d_ws;        // current node features (rep l)
    float* B      = A + nd;              // z / y2 buffer
    float* C      = B + nd;              // y1 buffer
    float* stats  = C + nd;              // 2*96: scale/shift scratch
    float* pooled = stats + 2 * DF;      // 5 * 64 * 96

    const int t4   = (int)((nd / 4 + 255) / 256);
    const int tEl  = (int)((nd + 255) / 256);
    const int scB  = (int)(((long)E * 24 + 255) / 256);
    const int gmB  = (M / 16 + 7) / 8;
    const int stB  = (M + 255) / 256;

    copy_f4<<<t4, 256, 0, stream>>>((float4*)A, (const float4*)h_in, (int)(nd / 4));

    for (int l = 0; l < NL; ++l) {
        pool_max<<<G, DF, 0, stream>>>(pooled + (long)l * G * DF, A, M, G);

        // z = h + segment_sum(h[src], dst)
        copy_f4<<<t4, 256, 0, stream>>>((float4*)B, (const float4*)A, (int)(nd / 4));
        scatter_add<<<scB, 256, 0, stream>>>(B, A, srcI, dstI, E);

        // y1 = z @ W1 + b1 ; BN ; ReLU
        zero_f32<<<1, 2 * DF, 0, stream>>>(stats, 2 * DF);
        gemm_wmma_f32<<<gmB, 256, 0, stream>>>(C, B, gW1 + (long)l * DF * DF, gb1 + l * DF, M);
        col_stats<<<stB, DF, 0, stream>>>(C, stats, M);
        finalize_stats<<<1, DF, 0, stream>>>(stats, gbn1g + l * DF, gbn1b + l * DF, M);
        bn_relu<<<tEl, 256, 0, stream>>>(C, stats, (int)nd);

        // y2 = y1 @ W2 + b2 ; outer BN ; PReLU -> h
        zero_f32<<<1, 2 * DF, 0, stream>>>(stats, 2 * DF);
        gemm_wmma_f32<<<gmB, 256, 0, stream>>>(B, C, gW2 + (long)l * DF * DF, gb2 + l * DF, M);
        col_stats<<<stB, DF, 0, stream>>>(B, stats, M);
        finalize_stats<<<1, DF, 0, stream>>>(stats, bng + l * DF, bnb + l * DF, M);
        bn_prelu<<<tEl, 256, 0, stream>>>(A, B, stats, pa, (int)nd);
    }
    pool_max<<<G, DF, 0, stream>>>(pooled + (long)NL * G * DF, A, M, G);

    head_gemm<<<NH, 128, 0, stream>>>((float*)d_out, pooled, linW, linb, G, NH);
}